// DynamicTopKRouter_83897891160585
// MI455X (gfx1250) — compile-verified
//
#include <hip/hip_runtime.h>
#include <hip/hip_bf16.h>

#define B_TOK 16384
#define HDIM  4096
#define NEXP  64
#define H4    1024
#define MAXK  8

typedef __attribute__((ext_vector_type(2))) float v2f;
typedef __attribute__((ext_vector_type(8))) float v8f;

// ---------------------------------------------------------------------------
// CDNA5 async memory->LDS copy (ASYNCcnt-tracked, no VGPR data staging).
// LDS generic pointers carry the wave-relative LDS byte offset in the low
// 32 bits, which is what the VDST VGPR of global_load_async_to_lds expects.
// ---------------------------------------------------------------------------
__device__ __forceinline__ void async_copy_b128(float* lds_dst, const float* gsrc) {
    uint32_t l = (uint32_t)(uintptr_t)lds_dst;
    uint64_t g = (uint64_t)(uintptr_t)gsrc;
    asm volatile("global_load_async_to_lds_b128 %0, %1, off"
                 :: "v"(l), "v"(g) : "memory");
}
__device__ __forceinline__ void async_wait0() {
    asm volatile("s_wait_asynccnt 0x0" ::: "memory");
}

// ---------------------------------------------------------------------------
// Kernel 0: zero workspace accumulators
// ws layout (floats): [0]=sigmoid sum, [1]=entropy sum, [2..65]=expert load, [66]=k (int)
// ---------------------------------------------------------------------------
__global__ void init_ws_kernel(float* __restrict__ ws) {
    int i = threadIdx.x;
    if (i < 66) ws[i] = 0.0f;
}

// ---------------------------------------------------------------------------
// Kernel 1: router_logits = X[B,H] @ Wr[E,H]^T  -> out[B,E]
// 256 threads = 8 waves; block covers 32 tokens x 64 experts; K staged in LDS
// via async-to-LDS; f32 WMMA 16x16x4.
// ---------------------------------------------------------------------------
__global__ __launch_bounds__(256) void router_gemm_kernel(
    const float* __restrict__ X, const float* __restrict__ Wr,
    float* __restrict__ out)
{
    constexpr int KC  = 64;
    constexpr int LDA = KC + 4;   // pad 4 floats: conflict-free strided v2f reads
    __shared__ float ldsA[32 * LDA];
    __shared__ float ldsB[64 * LDA];

    const int tid   = threadIdx.x;
    const int wave  = tid >> 5;
    const int lane  = tid & 31;
    const int half  = lane >> 4;
    const int l16   = lane & 15;
    const int mtile = wave >> 2;
    const int ntile = wave & 3;
    const int blockM = blockIdx.x * 32;

    // precomputed staging coordinates
    const int ar = (tid * 8) >> 6;          // A: 8 floats / thread (2 x b128)
    const int ac = (tid * 8) & 63;
    const int br = tid >> 2;                // B: 16 floats / thread (4 x b128)
    const int bc = (tid & 3) << 4;

    v8f acc = {};

    for (int k0 = 0; k0 < HDIM; k0 += KC) {
        // async stage A tile: 32 x KC
        async_copy_b128(&ldsA[ar * LDA + ac],
                        X + (size_t)(blockM + ar) * HDIM + k0 + ac);
        async_copy_b128(&ldsA[ar * LDA + ac + 4],
                        X + (size_t)(blockM + ar) * HDIM + k0 + ac + 4);
        // async stage B tile: 64 x KC
        #pragma unroll
        for (int q = 0; q < 4; ++q)
            async_copy_b128(&ldsB[br * LDA + bc + 4 * q],
                            Wr + (size_t)br * HDIM + k0 + bc + 4 * q);
        async_wait0();
        __syncthreads();

        #pragma unroll
        for (int kk = 0; kk < KC; kk += 4) {
            v2f a = *(const v2f*)&ldsA[(mtile * 16 + l16) * LDA + kk + 2 * half];
            v2f b = *(const v2f*)&ldsB[(ntile * 16 + l16) * LDA + kk + 2 * half];
            acc = __builtin_amdgcn_wmma_f32_16x16x4_f32(
                false, a, false, b, (short)0, acc, false, false);
        }
        __syncthreads();
    }

    const int row0 = blockM + mtile * 16;
    #pragma unroll
    for (int r = 0; r < 8; ++r) {
        int m = r + 8 * half;                       // C layout: VGPR r -> M=r / M=8+r
        out[(size_t)(row0 + m) * NEXP + ntile * 16 + l16] = acc[r];
    }
}

// ---------------------------------------------------------------------------
// Kernel 2: fused k-predictor.
// Per block: 16 tokens. h = relu(X@W1^T + b1) [16x1024] via WMMA (K=4096),
// immediately contracted with w2, sigmoid per token, atomic-add of the sum.
// 8 waves, each owns 128 of the 1024 hidden2 columns (8 N-tiles).
// W1 (16 MB) is L2-resident; next chunk is prefetched while WMMAs issue.
// ---------------------------------------------------------------------------
__global__ __launch_bounds__(256) void kpred_kernel(
    const float* __restrict__ X,  const float* __restrict__ W1,
    const float* __restrict__ b1, const float* __restrict__ w2,
    const float* __restrict__ b2, float* __restrict__ ws)
{
    constexpr int KC  = 64;
    constexpr int LDA = KC + 4;
    __shared__ float ldsA[16 * LDA];
    __shared__ float tokenSum[16];

    const int tid  = threadIdx.x;
    const int wave = tid >> 5;
    const int lane = tid & 31;
    const int half = lane >> 4;
    const int l16  = lane & 15;
    const int blockM = blockIdx.x * 16;

    if (tid < 16) tokenSum[tid] = 0.0f;

    const int ar = tid >> 4;                // A: 4 floats / thread (1 x b128)
    const int ac = (tid & 15) << 2;

    v8f zero = {};
    v8f acc[8];
    #pragma unroll
    for (int t = 0; t < 8; ++t) acc[t] = zero;

    for (int k0 = 0; k0 < HDIM; k0 += KC) {
        // async stage A tile: 16 x KC
        async_copy_b128(&ldsA[ar * LDA + ac],
                        X + (size_t)(blockM + ar) * HDIM + k0 + ac);
        async_wait0();
        __syncthreads();

        // prefetch next K-chunk of W1 (L2-resident; hides strided-load latency)
        if (k0 + KC < HDIM) {
            #pragma unroll
            for (int t = 0; t < 8; ++t) {
                int n = wave * 128 + t * 16 + l16;
                __builtin_prefetch(W1 + (size_t)n * HDIM + k0 + KC + 2 * half, 0, 1);
            }
        }

        #pragma unroll
        for (int kk = 0; kk < KC; kk += 4) {
            v2f a = *(const v2f*)&ldsA[l16 * LDA + kk + 2 * half];
            #pragma unroll
            for (int t = 0; t < 8; ++t) {
                int n = wave * 128 + t * 16 + l16;   // hidden2 column (W1 row)
                v2f b = *(const v2f*)(W1 + (size_t)n * HDIM + k0 + kk + 2 * half);
                acc[t] = __builtin_amdgcn_wmma_f32_16x16x4_f32(
                    false, a, false, b, (short)0, acc[t], false, false);
            }
        }
        __syncthreads();
    }

    // epilogue: relu(+b1) * w2, reduce over n, per-token sigmoid sum
    #pragma unroll
    for (int t = 0; t < 8; ++t) {
        int n = wave * 128 + t * 16 + l16;
        float bb  = b1[n];
        float w2n = w2[n];
        #pragma unroll
        for (int r = 0; r < 8; ++r) {
            float h = acc[t][r] + bb;
            h = h > 0.0f ? h : 0.0f;
            float v = h * w2n;
            // reduce across the 16 lanes of this half-wave (xor 1,2,4,8)
            v += __shfl_xor(v, 1, 32);
            v += __shfl_xor(v, 2, 32);
            v += __shfl_xor(v, 4, 32);
            v += __shfl_xor(v, 8, 32);
            if (l16 == 0) atomicAdd(&tokenSum[r + 8 * half], v);
        }
    }
    __syncthreads();

    if (tid == 0) {
        float s = 0.0f;
        float bias = b2[0];
        #pragma unroll
        for (int m = 0; m < 16; ++m) {
            float z = tokenSum[m] + bias;
            s += 1.0f / (1.0f + __expf(-z));
        }
        atomicAdd(&ws[0], s);
    }
}

// ---------------------------------------------------------------------------
// Kernel 3: derive dynamic k from mean sigmoid score
// ---------------------------------------------------------------------------
__global__ void compute_k_kernel(const float* __restrict__ ws_f, int* __restrict__ ws_k) {
    float mean = ws_f[0] / (float)B_TOK;
    int k = (int)(1.0f + mean * (float)(MAXK - 1));
    k = k < 1 ? 1 : (k > MAXK ? MAXK : k);
    *ws_k = k;
}

// ---------------------------------------------------------------------------
// Kernel 4: per-token softmax / top-k / expert-load / entropy.
// One wave per token (64 experts = 2 per lane on wave32).
// ---------------------------------------------------------------------------
__global__ __launch_bounds__(256) void topk_stats_kernel(
    const float* __restrict__ logits, const int* __restrict__ ws_k,
    float* __restrict__ out, float* __restrict__ ws)
{
    __shared__ float ldsLoad[NEXP];
    __shared__ float ldsEnt;

    const int tid  = threadIdx.x;
    const int wave = tid >> 5;
    const int lane = tid & 31;

    if (tid < NEXP) ldsLoad[tid] = 0.0f;
    if (tid == 0)   ldsEnt = 0.0f;
    __syncthreads();

    const int k = *ws_k;
    const int m = blockIdx.x * 8 + wave;
    const float* row = logits + (size_t)m * NEXP;

    float x0 = row[lane];
    float x1 = row[lane + 32];

    // full softmax over 64 experts
    float mx = fmaxf(x0, x1);
    #pragma unroll
    for (int d = 16; d >= 1; d >>= 1) mx = fmaxf(mx, __shfl_xor(mx, d, 32));
    float e0 = __expf(x0 - mx), e1 = __expf(x1 - mx);
    float s = e0 + e1;
    #pragma unroll
    for (int d = 16; d >= 1; d >>= 1) s += __shfl_xor(s, d, 32);
    float inv = 1.0f / s;
    float p0 = e0 * inv, p1 = e1 * inv;

    float ent = -(p0 * __logf(p0 + 1e-8f) + p1 * __logf(p1 + 1e-8f));
    #pragma unroll
    for (int d = 16; d >= 1; d >>= 1) ent += __shfl_xor(ent, d, 32);
    if (lane == 0) atomicAdd(&ldsEnt, ent);

    atomicAdd(&ldsLoad[lane], p0);
    atomicAdd(&ldsLoad[lane + 32], p1);

    // top-8 (use first k): iterative argmax, ties -> lower index (matches lax.top_k)
    float v0 = x0, v1 = x1;
    float topv[MAXK];
    int   topi[MAXK];
    #pragma unroll
    for (int j = 0; j < MAXK; ++j) {
        float bv; int bi;
        if (v0 >= v1) { bv = v0; bi = lane; } else { bv = v1; bi = lane + 32; }
        #pragma unroll
        for (int d = 16; d >= 1; d >>= 1) {
            float ov = __shfl_xor(bv, d, 32);
            int   oi = __shfl_xor(bi, d, 32);
            if (ov > bv || (ov == bv && oi < bi)) { bv = ov; bi = oi; }
        }
        topv[j] = bv; topi[j] = bi;
        if (bi == lane)      v0 = -__builtin_inff();
        if (bi == lane + 32) v1 = -__builtin_inff();
    }

    if (lane == 0) {
        float wm = topv[0];
        float ew[MAXK];
        #pragma unroll
        for (int j = 0; j < MAXK; ++j) ew[j] = __expf(topv[j] - wm);
        float es = 0.0f;
        for (int j = 0; j < k; ++j) es += ew[j];
        float r_ = 1.0f / es;
        size_t idx_off = (size_t)B_TOK * NEXP;
        size_t w_off   = idx_off + (size_t)B_TOK * (size_t)k;
        for (int j = 0; j < k; ++j) {
            out[idx_off + (size_t)m * k + j] = (float)topi[j];
            out[w_off   + (size_t)m * k + j] = ew[j] * r_;
        }
    }

    __syncthreads();
    if (tid < NEXP) atomicAdd(&ws[2 + tid], ldsLoad[tid]);
    if (tid == 0)   atomicAdd(&ws[1], ldsEnt);
}

// ---------------------------------------------------------------------------
// Kernel 5: finalize expert_load, unbiased variance, mean entropy
// ---------------------------------------------------------------------------
__global__ void finalize_kernel(const float* __restrict__ ws_f,
                                const int* __restrict__ ws_k,
                                float* __restrict__ out)
{
    __shared__ float sh[NEXP];
    const int e = threadIdx.x;     // 64 threads
    const int k = *ws_k;
    float le = ws_f[2 + e] / (float)B_TOK;
    size_t off = (size_t)B_TOK * NEXP + 2u * (size_t)B_TOK * (size_t)k;
    out[off + e] = le;
    sh[e] = le;
    __syncthreads();
    if (e == 0) {
        float mean = 0.0f;
        for (int i = 0; i < NEXP; ++i) mean += sh[i];
        mean /= (float)NEXP;
        float var = 0.0f;
        for (int i = 0; i < NEXP; ++i) { float d = sh[i] - mean; var += d * d; }
        var /= (float)(NEXP - 1);          // ddof=1
        out[off + NEXP]     = var;
        out[off + NEXP + 1] = ws_f[1] / (float)B_TOK;
    }
}

// ---------------------------------------------------------------------------
extern "C" void kernel_launch(void* const* d_in, const int* in_sizes, int n_in,
                              void* d_out, int out_size, void* d_ws, size_t ws_size,
                              hipStream_t stream) {
    const float* X  = (const float*)d_in[0];   // [B,H]
    const float* Wr = (const float*)d_in[1];   // [E,H]
    const float* W1 = (const float*)d_in[2];   // [H/4,H]
    const float* b1 = (const float*)d_in[3];   // [H/4]
    const float* w2 = (const float*)d_in[4];   // [1,H/4]
    const float* b2 = (const float*)d_in[5];   // [1]
    float* out = (float*)d_out;
    float* wsf = (float*)d_ws;
    int*   wsk = (int*)(wsf + 66);

    hipLaunchKernelGGL(init_ws_kernel, dim3(1), dim3(128), 0, stream, wsf);
    hipLaunchKernelGGL(router_gemm_kernel, dim3(B_TOK / 32), dim3(256), 0, stream,
                       X, Wr, out);
    hipLaunchKernelGGL(kpred_kernel, dim3(B_TOK / 16), dim3(256), 0, stream,
                       X, W1, b1, w2, b2, wsf);
    hipLaunchKernelGGL(compute_k_kernel, dim3(1), dim3(1), 0, stream, wsf, wsk);
    hipLaunchKernelGGL(topk_stats_kernel, dim3(B_TOK / 8), dim3(256), 0, stream,
                       out, wsk, out, wsf);
    hipLaunchKernelGGL(finalize_kernel, dim3(1), dim3(64), 0, stream, wsf, wsk, out);
}